// YGate_72773925863430
// MI455X (gfx1250) — compile-verified
//
#include <hip/hip_runtime.h>

typedef float f4  __attribute__((ext_vector_type(4)));
typedef int   v4i __attribute__((ext_vector_type(4)));
typedef v4i __attribute__((address_space(1))) g_v4i;   // global 4xi32
typedef v4i __attribute__((address_space(3))) l_v4i;   // LDS    4xi32

#define TPB   256          // 8 wave32 per block
#define TILES 8            // tiles (of TPB chunks) per block
// N = 2^24 floats per array; chunks = N/4 = 2^22; grid = 2^22 / (TPB*TILES) = 2048

#if defined(__AMDGCN__) && __has_builtin(__builtin_amdgcn_global_load_async_to_lds_b128)
#define YG_ASYNC 1
#else
#define YG_ASYNC 0
#endif

__device__ __forceinline__ void yg_wait_async_le2() {
#if __has_builtin(__builtin_amdgcn_s_wait_asynccnt)
    __builtin_amdgcn_s_wait_asynccnt(2);
#else
    asm volatile("s_wait_asynccnt 0x2" ::: "memory");
#endif
}
__device__ __forceinline__ void yg_wait_async_le0() {
#if __has_builtin(__builtin_amdgcn_s_wait_asynccnt)
    __builtin_amdgcn_s_wait_asynccnt(0);
#else
    asm volatile("s_wait_asynccnt 0x0" ::: "memory");
#endif
}

// out[i].re = s * x_im[i^2048]; out[i].im = -s * x_re[i^2048]; s = (i & 2048) ? -1 : +1
__global__ __launch_bounds__(TPB) void ygate_kernel(const float* __restrict__ xr,
                                                    const float* __restrict__ xi,
                                                    f4* __restrict__ out)
{
    const int tid    = threadIdx.x;
    const int block0 = blockIdx.x * (TPB * TILES);   // first chunk of this block

#if YG_ASYNC
    __shared__ f4 ldsR[2][TPB];
    __shared__ f4 ldsI[2][TPB];

    auto issue = [&](int t, int buf) {
        const int chunk = block0 + t * TPB + tid;
        const int j     = (chunk << 2) ^ 2048;       // partner element index (16B aligned)
        __builtin_amdgcn_global_load_async_to_lds_b128(
            (g_v4i*)(xr + j), (l_v4i*)&ldsR[buf][tid], 0, 0);
        __builtin_amdgcn_global_load_async_to_lds_b128(
            (g_v4i*)(xi + j), (l_v4i*)&ldsI[buf][tid], 0, 0);
    };

    issue(0, 0);
    for (int t = 0; t < TILES; ++t) {
        if (t + 1 < TILES) {
            issue(t + 1, (t + 1) & 1);
            yg_wait_async_le2();   // async loads complete in order -> tile t's 2 loads done
        } else {
            yg_wait_async_le0();
        }
        const int buf   = t & 1;
        const f4  vr    = ldsR[buf][tid];
        const f4  vi    = ldsI[buf][tid];
        const int chunk = block0 + t * TPB + tid;
        const float s   = ((chunk << 2) & 2048) ? -1.0f : 1.0f;

        f4 lo, hi;  // interleaved complex64: (re, im) pairs
        lo.x = s * vi.x;  lo.y = -s * vr.x;
        lo.z = s * vi.y;  lo.w = -s * vr.y;
        hi.x = s * vi.z;  hi.y = -s * vr.z;
        hi.z = s * vi.w;  hi.w = -s * vr.w;
        __builtin_nontemporal_store(lo, &out[chunk * 2 + 0]);
        __builtin_nontemporal_store(hi, &out[chunk * 2 + 1]);
    }
#else
    for (int t = 0; t < TILES; ++t) {
        const int chunk = block0 + t * TPB + tid;
        const int j     = (chunk << 2) ^ 2048;
        const f4  vr    = __builtin_nontemporal_load((const f4*)(xr + j));
        const f4  vi    = __builtin_nontemporal_load((const f4*)(xi + j));
        const float s   = ((chunk << 2) & 2048) ? -1.0f : 1.0f;

        f4 lo, hi;
        lo.x = s * vi.x;  lo.y = -s * vr.x;
        lo.z = s * vi.y;  lo.w = -s * vr.y;
        hi.x = s * vi.z;  hi.y = -s * vr.z;
        hi.z = s * vi.w;  hi.w = -s * vr.w;
        __builtin_nontemporal_store(lo, &out[chunk * 2 + 0]);
        __builtin_nontemporal_store(hi, &out[chunk * 2 + 1]);
    }
#endif
}

extern "C" void kernel_launch(void* const* d_in, const int* in_sizes, int n_in,
                              void* d_out, int out_size, void* d_ws, size_t ws_size,
                              hipStream_t stream) {
    (void)n_in; (void)out_size; (void)d_ws; (void)ws_size;
    const float* xr = (const float*)d_in[0];   // x_real, 2^24 floats
    const float* xi = (const float*)d_in[1];   // x_imag, 2^24 floats
    f4* out = (f4*)d_out;                      // complex64 interleaved -> float4 pairs

    const int n       = in_sizes[0];           // 2^24
    const int nchunks = n / 4;                 // 2^22 float4 chunks
    const int grid    = nchunks / (TPB * TILES); // 2048 blocks, exact cover

    ygate_kernel<<<grid, TPB, 0, stream>>>(xr, xi, out);
}